// APPM_996432413602
// MI455X (gfx1250) — compile-verified
//
#include <hip/hip_runtime.h>
#include <hip/hip_bf16.h>

typedef float v2f __attribute__((ext_vector_type(2)));
typedef float v8f __attribute__((ext_vector_type(8)));

#define NB 64
#define NC 2048
#define HW 196          // 14*14
#define NTILES 13       // 13*16 = 208 >= 196
#define TILEPAD 208
#define NWIN 917

__constant__ int c_rh[13]   = {4,3,5,6,5,7,8,6,10,7,9,7,10};
__constant__ int c_rw[13]   = {4,5,3,6,7,5,8,10,6,9,7,10,7};
__constant__ int c_woff[14] = {0,121,241,361,442,522,602,651,696,741,789,837,877,917};

// ---------------------------------------------------------------------------
// Kernel 1: channel-sum via V_WMMA_F32_16X16X4_F32 with A = ones(16x4).
// D = ones * B + D  => every row of D accumulates the K-sum of B, i.e. the
// 4-channel partial sum per spatial column. One wave per (tile, chunk, batch).
// ---------------------------------------------------------------------------
__global__ __launch_bounds__(32)
void appm_chansum_wmma(const float* __restrict__ x, float* __restrict__ partial,
                       int nchunk) {
  const int tile  = blockIdx.x;        // 0..12  -> spatial 16-column tile
  const int chunk = blockIdx.y;        // 0..nchunk-1 -> channel slice
  const int b     = blockIdx.z;        // batch
  const int cper  = NC / nchunk;
  const int lane  = threadIdx.x;
  const int n     = lane & 15;         // B-matrix column (spatial)
  const int h     = lane >> 4;         // half-wave: selects channel pair
  const int hw    = tile * 16 + n;
  const int hwc   = hw < HW ? hw : (HW - 1);   // clamp, mask below
  const float msk = hw < HW ? 1.0f : 0.0f;

  const float* p = x + ((size_t)b * NC + (size_t)chunk * cper + 2 * h) * HW + hwc;

  v2f a; a.x = 1.0f; a.y = 1.0f;       // ones A-matrix (16x4)
  v8f acc = {};

  const int steps = cper >> 2;         // 4 channels per WMMA
#pragma unroll 4
  for (int it = 0; it < steps; ++it) {
    v2f bv;
    bv.x = msk * p[0];                 // channel c0 + 2h
    bv.y = msk * p[HW];                // channel c0 + 2h + 1
    p += 4 * HW;
    acc = __builtin_amdgcn_wmma_f32_16x16x4_f32(
        false, a, false, bv, (short)0, acc, false, false);
  }

  if (lane < 16) {                     // D row 0, N = lane (all rows equal)
    partial[((size_t)b * nchunk + chunk) * TILEPAD + tile * 16 + n] = acc[0];
  }
}

// ---------------------------------------------------------------------------
// Kernel 2: per-batch — reduce partials, SAT, window scores, greedy NMS.
// ---------------------------------------------------------------------------
__global__ __launch_bounds__(256)
void appm_score_nms(const float* __restrict__ partial, float* __restrict__ out,
                    int nchunk) {
  __shared__ float s[HW];
  __shared__ float sat[15 * 15];
  __shared__ float wsc[NWIN];
  __shared__ float det[NWIN];
  __shared__ float cx1[NWIN], cy1[NWIN], cx2[NWIN], cy2[NWIN], car[NWIN];
  __shared__ float rv[256];
  __shared__ int   ri[256];

  const int b   = blockIdx.x;
  const int tid = threadIdx.x;

  // A) deterministic reduce over channel chunks
  for (int hw = tid; hw < HW; hw += 256) {
    float acc = 0.f;
    for (int c = 0; c < nchunk; ++c)
      acc += partial[((size_t)b * nchunk + c) * TILEPAD + hw];
    s[hw] = acc;
  }
  for (int i = tid; i < 15 * 15; i += 256) sat[i] = 0.f;
  __syncthreads();

  // B) summed-area table: row prefix then column prefix
  if (tid < 14) {
    float run = 0.f;
    for (int j = 0; j < 14; ++j) {
      run += s[tid * 14 + j];
      sat[(tid + 1) * 15 + (j + 1)] = run;
    }
  }
  __syncthreads();
  if (tid < 14) {
    float run = 0.f;
    for (int i = 1; i <= 14; ++i) {
      run += sat[i * 15 + (tid + 1)];
      sat[i * 15 + (tid + 1)] = run;
    }
  }
  __syncthreads();

  // C) window scores + box coords (replicates ComputeCoordinate exactly:
  //    x2/y2 use the UNclamped x1/y1; clamp applied after)
  for (int w = tid; w < NWIN; w += 256) {
    int r = 0;
    while (w >= c_woff[r + 1]) ++r;
    const int local = w - c_woff[r];
    const int rh = c_rh[r], rw = c_rw[r];
    const int ww = 15 - rw;
    const int xi = local / ww;
    const int yi = local - xi * ww;

    const float sum = sat[(xi + rh) * 15 + (yi + rw)] - sat[xi * 15 + (yi + rw)]
                    - sat[(xi + rh) * 15 + yi]        + sat[xi * 15 + yi];
    const float score = sum / (float)(rh * rw);
    wsc[w] = score;
    det[w] = score;
    out[2 * NB * 7 + (size_t)b * NWIN + w] = score;   // window_scores

    const float x1 = (float)(xi * 32 - 1);
    const float y1 = (float)(yi * 32 - 1);
    const float x2 = x1 + (float)(rh * 32);
    const float y2 = y1 + (float)(rw * 32);
    const float x1c = fmaxf(x1, 0.f);
    const float y1c = fmaxf(y1, 0.f);
    cx1[w] = x1c; cy1[w] = y1c; cx2[w] = x2; cy2[w] = y2;
    car[w] = (x2 - x1c + 1.f) * (y2 - y1c + 1.f);
  }
  __syncthreads();

  // D) greedy NMS: groups [0,361) [361,602) [602,917); select 2,3,2
  const int gb[4]   = {0, 361, 602, 917};
  const int nsel[3] = {2, 3, 2};
  int slot = 0;
  for (int g = 0; g < 3; ++g) {
    const int lo = gb[g], hi = gb[g + 1];
    for (int k = 0; k < nsel[g]; ++k) {
      // block-wide argmax, lowest index wins ties (matches jnp.argmax)
      float best = -__builtin_inff();
      int   bi   = 0x7fffffff;
      for (int w = lo + tid; w < hi; w += 256) {
        const float v = det[w];
        if (v > best || (v == best && w < bi)) { best = v; bi = w; }
      }
      rv[tid] = best; ri[tid] = bi;
      __syncthreads();
      for (int off = 128; off > 0; off >>= 1) {
        if (tid < off) {
          const float ov = rv[tid + off];
          const int   oi = ri[tid + off];
          if (ov > rv[tid] || (ov == rv[tid] && oi < ri[tid])) {
            rv[tid] = ov; ri[tid] = oi;
          }
        }
        __syncthreads();
      }
      const int sel = ri[0];
      if (tid == 0) {
        out[(size_t)b * 7 + slot]          = (float)sel;   // proposalN_indices
        out[NB * 7 + (size_t)b * 7 + slot] = wsc[sel];     // original scores
      }
      const float bx1 = cx1[sel], by1 = cy1[sel];
      const float bx2 = cx2[sel], by2 = cy2[sel];
      const float ba  = car[sel];
      for (int w = lo + tid; w < hi; w += 256) {
        const float xx1 = fmaxf(cx1[w], bx1);
        const float yy1 = fmaxf(cy1[w], by1);
        const float xx2 = fminf(cx2[w], bx2);
        const float yy2 = fminf(cy2[w], by2);
        const float lw = xx2 - xx1 + 1.f;
        const float lh = yy2 - yy1 + 1.f;
        const float inter = (lw < 0.f || lh < 0.f) ? 0.f : lw * lh;
        const float iou = inter / (car[w] + ba - inter);
        if (iou > 0.25f || w == sel) det[w] = -__builtin_inff();
      }
      __syncthreads();
      ++slot;
    }
  }
}

extern "C" void kernel_launch(void* const* d_in, const int* in_sizes, int n_in,
                              void* d_out, int out_size, void* d_ws, size_t ws_size,
                              hipStream_t stream) {
  (void)in_sizes; (void)n_in; (void)out_size;
  const float* x  = (const float*)d_in[0];
  float* out      = (float*)d_out;
  float* partial  = (float*)d_ws;

  int nchunk = 8;   // 8 channel slices of 256 -> 6656 waves for latency hiding
  if (ws_size < (size_t)NB * 8 * TILEPAD * sizeof(float)) nchunk = 1;

  dim3 g1(NTILES, nchunk, NB);
  appm_chansum_wmma<<<g1, 32, 0, stream>>>(x, partial, nchunk);
  appm_score_nms<<<NB, 256, 0, stream>>>(partial, out, nchunk);
}